// CVRNNLayer_27462020890971
// MI455X (gfx1250) — compile-verified
//
#include <hip/hip_runtime.h>

// CVRNN: x_{t+1} = i*omega .* x_t + B @ x_t   (complex64), NT=1024 steps, N=4096
// d_in[0] = omega (float, N), d_in[1] = B (complex64 -> 2*N*N floats, interleaved),
// d_in[2] = x0 (complex64 -> 2*N floats). d_out = (NT, N) complex64 -> NT*2*N floats.

#define CN 4096
#define CNT 1024
#define ROWS_PER_BLOCK 16
#define WAVES_PER_BLOCK 8
#define THREADS (WAVES_PER_BLOCK * 32)
#define K_PER_WAVE (CN / WAVES_PER_BLOCK)   // 512
#define X_FLOATS_PER_WAVE (2 * CN / WAVES_PER_BLOCK)  // 1024 floats = 4KB slice

typedef __attribute__((ext_vector_type(2))) float v2f;
typedef __attribute__((ext_vector_type(8))) float v8f;
typedef __attribute__((ext_vector_type(4))) unsigned int v4u;
typedef __attribute__((ext_vector_type(8))) int v8i_;
typedef __attribute__((ext_vector_type(4))) int v4i_;

__global__ __launch_bounds__(THREADS)
void cvrnn_step_kernel(const float* __restrict__ Bm,     // 2*N*N floats (re,im interleaved, row-major)
                       const float* __restrict__ omega,  // N floats
                       const float* __restrict__ xin,    // 2*N floats (row t of history)
                       float* __restrict__ xout)         // 2*N floats (row t+1)
{
    __shared__ float xs[2 * CN];                          // 32 KB: x_t (re,im interleaved)
    __shared__ float red[WAVES_PER_BLOCK][32][16];        // 16 KB: per-wave partial tiles
    __shared__ float sum[32][16];                         //  2 KB: reduced tiles

    const int tid  = threadIdx.x;
    const int wave = tid >> 5;
    const int lane = tid & 31;

    // ---- stage x_t into LDS via the Tensor Data Mover: one 4KB 1-D tile per wave ----
    {
        const uint64_t gaddr = (uint64_t)(uintptr_t)(xin + (size_t)wave * X_FLOATS_PER_WAVE);
        const uint32_t lds_base =
            ((uint32_t)(uintptr_t)(void*)&xs[0]) + (uint32_t)(wave * X_FLOATS_PER_WAVE * 4);
        const uint32_t elems = X_FLOATS_PER_WAVE;         // 1024 f32 elements

        // D# group 0: count=1 | lds_addr | global_addr[56:0] | type=2 ("image")
        v4u g0;
        g0[0] = 1u;
        g0[1] = lds_base;
        g0[2] = (uint32_t)gaddr;
        g0[3] = ((uint32_t)(gaddr >> 32) & 0x01ffffffu) | (2u << 30);

        // D# group 1: mask=0, data_size=2 (4B), tensor_dim0=tile_dim0=1024, dims>=1 unused
        v8i_ g1;
        g1[0] = (int)(2u << 16);                          // data_size = 4 bytes
        g1[1] = (int)((elems & 0xffffu) << 16);           // tensor_dim0[15:0]  (bits 63:48)
        g1[2] = (int)(elems >> 16);                       // tensor_dim0[31:16] (bits 79:64)
        g1[3] = (int)((elems & 0xffffu) << 16);           // tile_dim0          (bits 127:112)
        g1[4] = 0;                                        // tile_dim1=0, tile_dim2=0
        g1[5] = (int)elems;                               // tensor_dim0_stride[31:0]
        g1[6] = 0;
        g1[7] = 0;

        v4i_ gz4 = {0, 0, 0, 0};
        v8i_ gz8 = {0, 0, 0, 0, 0, 0, 0, 0};
        // amdgpu-toolchain (clang-23 / therock-10.0) 6-arg form
        __builtin_amdgcn_tensor_load_to_lds(g0, g1, gz4, gz4, gz8, 0);
        __builtin_amdgcn_s_wait_tensorcnt(0);             // own slice done before barrier
    }
    __syncthreads();                                      // all 8 slices visible

    const int m     = lane & 15;               // row-in-tile (A) / column N (B-operand)
    const int halfK = (lane < 16) ? 0 : 2;     // K pair handled by this lane half
    const int row   = blockIdx.x * ROWS_PER_BLOCK + m;
    const int comp  = m & 1;                   // even cols carry xr, odd cols carry xi

    v8f cr = {};   // accumulates Br@xr (col0) / Br@xi (col1)
    v8f ci = {};   // accumulates Bi@xr (col0) / Bi@xi (col1)

    const int kbase0 = wave * K_PER_WAVE;
    // this lane's B stream: complex element (row, kbase0 + halfK), 16B-aligned
    const float* bptr = Bm + (((size_t)row * CN + (size_t)(kbase0 + halfK)) * 2);

    // unroll 8: up to 8 b128 loads in flight per wave (MLP covers HBM/L2 latency;
    // no prefetch -- SE-scope global_prefetch only reaches GL2 where B already lives)
    #pragma unroll 8
    for (int it = 0; it < K_PER_WAVE / 4; ++it) {
        const int kb = kbase0 + it * 4;
        // {re(k0), im(k0), re(k0+1), im(k0+1)} for K = halfK, halfK+1 of this chunk
        float4 bq = *(const float4*)bptr;
        bptr += 8;                                   // advance 4 complex = 8 floats

        v2f Ar = { bq.x, bq.z };                     // Br tile slots (VGPR0, VGPR1)
        v2f Ai = { bq.y, bq.w };                     // Bi tile slots

        // B operand (4x16): VGPR0 -> K=halfK, VGPR1 -> K=halfK+1, col m parity picks re/im
        const int k0 = kb + halfK;
        v2f Bo = { xs[k0 * 2 + comp], xs[(k0 + 1) * 2 + comp] };

        cr = __builtin_amdgcn_wmma_f32_16x16x4_f32(false, Ar, false, Bo,
                                                   (short)0, cr, false, false);
        ci = __builtin_amdgcn_wmma_f32_16x16x4_f32(false, Ai, false, Bo,
                                                   (short)0, ci, false, false);
    }

    // ---- per-wave partials -> LDS (b128 stores, 16 consecutive floats per lane) ----
    {
        float4* rp = (float4*)&red[wave][lane][0];
        rp[0] = make_float4(cr[0], cr[1], cr[2], cr[3]);
        rp[1] = make_float4(cr[4], cr[5], cr[6], cr[7]);
        rp[2] = make_float4(ci[0], ci[1], ci[2], ci[3]);
        rp[3] = make_float4(ci[4], ci[5], ci[6], ci[7]);
    }
    __syncthreads();

    // ---- deterministic cross-wave reduction (fixed order; no float atomics) ----
    {
        const int l  = tid >> 3;            // 0..31 (lane slot)
        const int kk = (tid & 7) * 2;       // 0..14 (pair of tile entries)
        float s0 = 0.f, s1 = 0.f;
        #pragma unroll
        for (int w = 0; w < WAVES_PER_BLOCK; ++w) {
            s0 += red[w][l][kk];
            s1 += red[w][l][kk + 1];
        }
        sum[l][kk] = s0;
        sum[l][kk + 1] = s1;
    }
    __syncthreads();

    // ---- final combine: y = (Br@xr - Bi@xi - w*xi) + i (Br@xi + Bi@xr + w*xr) ----
    if (tid < 16) {
        const int M  = tid;
        const int lr = (M < 8) ? 0 : 16;    // lane holding D[M][0] per C/D layout
        const int v  = (M < 8) ? M : M - 8; // VGPR index within accumulator
        const float crm0 = sum[lr][v];          // (Br@xr)[M]
        const float crm1 = sum[lr + 1][v];      // (Br@xi)[M]
        const float cim0 = sum[lr][8 + v];      // (Bi@xr)[M]
        const float cim1 = sum[lr + 1][8 + v];  // (Bi@xi)[M]

        const int r  = blockIdx.x * ROWS_PER_BLOCK + M;
        const float xr = xs[r * 2];
        const float xi = xs[r * 2 + 1];
        const float om = omega[r];

        const float yr = crm0 - cim1 - om * xi;
        const float yi = crm1 + cim0 + om * xr;
        *(float2*)(xout + r * 2) = make_float2(yr, yi);
    }
}

__global__ void cvrnn_copy_x0(const float* __restrict__ x0, float* __restrict__ out0)
{
    const int i = blockIdx.x * blockDim.x + threadIdx.x;
    if (i < 2 * CN) out0[i] = x0[i];
}

extern "C" void kernel_launch(void* const* d_in, const int* in_sizes, int n_in,
                              void* d_out, int out_size, void* d_ws, size_t ws_size,
                              hipStream_t stream)
{
    (void)in_sizes; (void)n_in; (void)out_size; (void)d_ws; (void)ws_size;
    const float* omega = (const float*)d_in[0];
    const float* Bm    = (const float*)d_in[1];
    const float* x0    = (const float*)d_in[2];
    float* out = (float*)d_out;

    // history row 0 = x0
    cvrnn_copy_x0<<<(2 * CN + 255) / 256, 256, 0, stream>>>(x0, out);

    // 1023 dependent GEMV steps; B stays hot in the 192MB L2 after step 0
    for (int t = 0; t < CNT - 1; ++t) {
        cvrnn_step_kernel<<<CN / ROWS_PER_BLOCK, THREADS, 0, stream>>>(
            Bm, omega,
            out + (size_t)t * 2 * CN,
            out + (size_t)(t + 1) * 2 * CN);
    }
}